// SpectralS6Block_55731495633373
// MI455X (gfx1250) — compile-verified
//
#include <hip/hip_runtime.h>
#include <hip/hip_bf16.h>

typedef __attribute__((ext_vector_type(2))) float v2f;
typedef __attribute__((ext_vector_type(8))) float v8f;

#define B_SZ     2
#define SEQLEN   1024
#define D_MODEL  768
#define D_INNER  1536
#define D_STATE  64
#define DT_RANK  48
#define KST      32
#define D_XDBL   (DT_RANK + 2 * D_STATE)   /* 176 */
#define MROWS    (B_SZ * SEQLEN)           /* 2048 */

static __device__ __forceinline__ v8f wmma_f32(v2f a, v2f b, v8f c) {
    return __builtin_amdgcn_wmma_f32_16x16x4_f32(
        /*neg_a=*/false, a, /*neg_b=*/false, b,
        /*c_mod=*/(short)0, c, /*reuse_a=*/false, /*reuse_b=*/false);
}

// ---------------------------------------------------------------------------
// C[M, nOffset : nOffset+tilesN*16*NSUB] = A[M,K] * B[N,K]^T (+bias)
// via V_WMMA_F32_16X16X4_F32.  One wave32 computes a 16(M) x (16*NSUB)(N)
// strip; NSUB is a template constant so the inner loop is straight-line:
// one A-fragment load feeds NSUB WMMAs per K-step, no EXEC manipulation.
// A-frag: lane holds row m=lane&15, float2 at k + 2*(lane>>4).
// B-frag: lane holds W row n=lane&15 of the sub-tile, same k slice.
// D layout: vgpr r -> (m = r + 8*(lane>>4), n = lane&15).
// ---------------------------------------------------------------------------
template <int NSUB>
__global__ __launch_bounds__(256) void s6_gemm_nt_wmma(
    const float* __restrict__ A, int lda,
    const float* __restrict__ Bw, int ldb,
    const float* __restrict__ bias,
    float* __restrict__ C, int ldc,
    int M, int tilesN, int nOffset, int Kd)
{
    const int wave = threadIdx.x >> 5;
    const int lane = threadIdx.x & 31;
    const long nTiles = (long)(M >> 4) * tilesN;
    const long tileId = (long)blockIdx.x * 8 + wave;    // uniform per wave
    if (tileId >= nTiles) return;                       // wave-uniform exit

    const int tm    = (int)(tileId / tilesN);
    const int tn    = (int)(tileId % tilesN);
    const int nBase = nOffset + tn * (NSUB << 4);

    const int half = lane >> 4;                         // 0 or 1
    const int kh   = half << 1;                         // 0 or 2
    const int l15  = lane & 15;
    const float* ap = A  + (long)((tm << 4) + l15) * lda + kh;
    const float* bp = Bw + (long)(nBase + l15) * ldb + kh;
    const long bstep = (long)ldb << 4;                  // 16 rows of W

    v8f acc[NSUB] = {};
    #pragma unroll 2
    for (int k = 0; k < Kd; k += 4) {
        __builtin_prefetch(ap + k + 64, 0, 3);
        __builtin_prefetch(bp + k + 64, 0, 3);
        const v2f a = *(const v2f*)(ap + k);
        #pragma unroll
        for (int s = 0; s < NSUB; ++s) {
            v2f b = *(const v2f*)(bp + (long)s * bstep + k);
            acc[s] = wmma_f32(a, b, acc[s]);
        }
    }

    const int rowBase = (tm << 4) + (half << 3);
    #pragma unroll
    for (int s = 0; s < NSUB; ++s) {
        const int col = nBase + (s << 4) + l15;
        const float bv = bias ? bias[col] : 0.0f;
        #pragma unroll
        for (int r = 0; r < 8; ++r)
            C[(long)(rowBase + r) * ldc + col] = acc[s][r] + bv;
    }
}

// ---------------------------------------------------------------------------
// Depthwise causal conv (width 4) + bias + SiLU.  x = xz[..., :D_INNER].
// ---------------------------------------------------------------------------
__global__ __launch_bounds__(256) void s6_conv_silu(
    const float* __restrict__ xz, const float* __restrict__ cw,
    const float* __restrict__ cb, float* __restrict__ u)
{
    long idx = (long)blockIdx.x * blockDim.x + threadIdx.x;
    const long total = (long)MROWS * D_INNER;
    if (idx >= total) return;
    const int c   = (int)(idx % D_INNER);
    const long bl = idx / D_INNER;          // b*SEQLEN + l
    const int l   = (int)(bl % SEQLEN);

    float acc = cb[c];
    #pragma unroll
    for (int j = 0; j < 4; ++j) {
        const int ls = l - 3 + j;
        if (ls >= 0)
            acc += xz[(bl - 3 + j) * (2 * D_INNER) + c] * cw[c * 4 + j];
    }
    u[idx] = acc / (1.0f + __expf(-acc));   // SiLU
}

// ---------------------------------------------------------------------------
// delta = softplus(dt + b_dt)  (b_dt already added once inside the GEMM;
// reference adds it twice).  In-place on the dt buffer.
// ---------------------------------------------------------------------------
__global__ __launch_bounds__(256) void s6_softplus(
    float* __restrict__ dt, const float* __restrict__ b_dt)
{
    long idx = (long)blockIdx.x * blockDim.x + threadIdx.x;
    const long total = (long)MROWS * D_INNER;
    if (idx >= total) return;
    const float v = dt[idx] + b_dt[idx % D_INNER];
    dt[idx] = (v > 20.0f) ? v : log1pf(__expf(v));
}

// ---------------------------------------------------------------------------
// Selective scan: one wave32 per (b, channel); lane = state index n (K=32).
// s_t = exp(delta*A)*s_{t-1} + delta*u*B ;  y = <s,C> + u*D ;  y *= silu(z)
// Cross-lane dot product via wave32 shfl_xor butterfly (warpSize==32).
// ---------------------------------------------------------------------------
__global__ __launch_bounds__(256) void s6_scan(
    const float* __restrict__ delta, const float* __restrict__ u,
    const float* __restrict__ xdbl,  const float* __restrict__ xz,
    const float* __restrict__ A_log, const float* __restrict__ Dp,
    float* __restrict__ y)
{
    const int wave = threadIdx.x >> 5;
    const int lane = threadIdx.x & 31;
    const int chan = blockIdx.x * 8 + wave;        // 0 .. B_SZ*D_INNER-1
    if (chan >= B_SZ * D_INNER) return;
    const int b = chan / D_INNER;
    const int c = chan % D_INNER;

    const float a    = -__expf(A_log[c * D_STATE + lane]);   // A_k[c, n]
    const float dpar = Dp[c];
    float s = 0.0f;

    for (int t = 0; t < SEQLEN; ++t) {
        const long row = (long)b * SEQLEN + t;
        const float dv = delta[row * D_INNER + c];
        const float uv = u    [row * D_INNER + c];
        const float bv = xdbl[row * D_XDBL + DT_RANK + lane];
        const float cv = xdbl[row * D_XDBL + DT_RANK + D_STATE + lane];

        s = __expf(dv * a) * s + dv * uv * bv;

        float p = s * cv;
        #pragma unroll
        for (int off = 16; off > 0; off >>= 1)
            p += __shfl_xor(p, off, 32);

        if (lane == 0) {
            const float zv = xz[row * (2 * D_INNER) + D_INNER + c];
            const float sz = zv / (1.0f + __expf(-zv));
            y[row * D_INNER + c] = (p + uv * dpar) * sz;
        }
    }
}

// ---------------------------------------------------------------------------
static void launch_gemm_nt(const float* A, int lda, const float* Bw, int ldb,
                           const float* bias, float* C, int ldc,
                           int M, int N, int Kd, hipStream_t stream)
{
    const int n64 = N >> 6;            // full 64-wide strips
    const int rem = (N & 63) >> 4;     // leftover 16-col subtiles (0..3)
    if (n64 > 0) {
        const int tiles = (M >> 4) * n64;
        s6_gemm_nt_wmma<4><<<(tiles + 7) / 8, 256, 0, stream>>>(
            A, lda, Bw, ldb, bias, C, ldc, M, n64, 0, Kd);
    }
    const int tilesR = (M >> 4);
    if (rem == 3)
        s6_gemm_nt_wmma<3><<<(tilesR + 7) / 8, 256, 0, stream>>>(
            A, lda, Bw, ldb, bias, C, ldc, M, 1, n64 << 6, Kd);
    else if (rem == 2)
        s6_gemm_nt_wmma<2><<<(tilesR + 7) / 8, 256, 0, stream>>>(
            A, lda, Bw, ldb, bias, C, ldc, M, 1, n64 << 6, Kd);
    else if (rem == 1)
        s6_gemm_nt_wmma<1><<<(tilesR + 7) / 8, 256, 0, stream>>>(
            A, lda, Bw, ldb, bias, C, ldc, M, 1, n64 << 6, Kd);
}

extern "C" void kernel_launch(void* const* d_in, const int* in_sizes, int n_in,
                              void* d_out, int out_size, void* d_ws, size_t ws_size,
                              hipStream_t stream) {
    const float* hs     = (const float*)d_in[0];
    const float* W_in   = (const float*)d_in[1];
    const float* conv_w = (const float*)d_in[2];
    const float* conv_b = (const float*)d_in[3];
    const float* W_x    = (const float*)d_in[4];
    const float* W_dt   = (const float*)d_in[5];
    const float* b_dt   = (const float*)d_in[6];
    const float* A_log  = (const float*)d_in[7];
    const float* D_par  = (const float*)d_in[8];
    const float* W_out  = (const float*)d_in[9];
    float* out = (float*)d_out;

    // Workspace carve-up (floats)
    char* ws = (char*)d_ws;
    float* xz    = (float*)(ws);                                   // 2048 x 3072
    float* u     = (float*)(ws + (size_t)MROWS * 2 * D_INNER * 4); // 2048 x 1536
    float* xdbl  = (float*)((char*)u     + (size_t)MROWS * D_INNER * 4); // 2048 x 176
    float* delta = (float*)((char*)xdbl  + (size_t)MROWS * D_XDBL * 4);  // 2048 x 1536
    float* ybuf  = (float*)((char*)delta + (size_t)MROWS * D_INNER * 4); // 2048 x 1536

    const long elems = (long)MROWS * D_INNER;
    const int ewBlocks = (int)((elems + 255) / 256);

    // 1) xz = hs @ W_in^T
    launch_gemm_nt(hs, D_MODEL, W_in, D_MODEL, nullptr, xz, 2 * D_INNER,
                   MROWS, 2 * D_INNER, D_MODEL, stream);

    // 2) depthwise conv + SiLU -> u
    s6_conv_silu<<<ewBlocks, 256, 0, stream>>>(xz, conv_w, conv_b, u);

    // 3) x_dbl = u @ W_x^T   (N = 176 -> one NSUB=4 pass + one NSUB=3 pass)
    launch_gemm_nt(u, D_INNER, W_x, D_INNER, nullptr, xdbl, D_XDBL,
                   MROWS, D_XDBL, D_INNER, stream);

    // 4) dt = dt_r @ W_dt^T + b_dt   (dt_r = x_dbl[..., :48], lda = 176)
    launch_gemm_nt(xdbl, D_XDBL, W_dt, DT_RANK, b_dt, delta, D_INNER,
                   MROWS, D_INNER, DT_RANK, stream);

    // 5) delta = softplus(dt + b_dt)   (second b_dt add, matching reference)
    s6_softplus<<<ewBlocks, 256, 0, stream>>>(delta, b_dt);

    // 6) selective scan + D skip + SiLU(z) gate -> ybuf
    s6_scan<<<(B_SZ * D_INNER + 7) / 8, 256, 0, stream>>>(
        delta, u, xdbl, xz, A_log, D_par, ybuf);

    // 7) out = ybuf @ W_out^T
    launch_gemm_nt(ybuf, D_INNER, W_out, D_INNER, nullptr, out, D_MODEL,
                   MROWS, D_MODEL, D_INNER, stream);
}